// MemN2N_43508018708770
// MI455X (gfx1250) — compile-verified
//
#include <hip/hip_runtime.h>

typedef float v2f __attribute__((ext_vector_type(2)));
typedef float v8f __attribute__((ext_vector_type(8)));

#define BB 64
#define SS 4096
#define DD 256
#define CH 16                    // split-K chunks over S
#define ROWS_PER_CH (SS / CH)    // 256 rows per chunk

// ---------------------------------------------------------------------------
// Kernel 1: lengths[b] = popcount(mask[b,:])
// ---------------------------------------------------------------------------
__global__ void k_lengths(const unsigned char* __restrict__ mask,
                          int* __restrict__ lengths) {
    __shared__ int sred[256];
    const int b = blockIdx.x;
    const int t = threadIdx.x;
    int cnt = 0;
    for (int i = t; i < SS; i += 256)
        cnt += (mask[(size_t)b * SS + i] != 0) ? 1 : 0;
    sred[t] = cnt;
    __syncthreads();
    for (int off = 128; off > 0; off >>= 1) {
        if (t < off) sred[t] += sred[t + off];
        __syncthreads();
    }
    if (t == 0) lengths[b] = sred[0];
}

// ---------------------------------------------------------------------------
// Kernel 2: partial masked row-sums (split-K over S, deterministic)
// partials[b][ch][d] = sum_{s in chunk, s < len[b]-1} sentences[b,s,d]
// ---------------------------------------------------------------------------
__global__ void k_partial(const float* __restrict__ sent,
                          const int* __restrict__ lengths,
                          float* __restrict__ partials) {
    const int ch = blockIdx.x;
    const int b  = blockIdx.y;
    const int d  = threadIdx.x;          // 0..255 == DD
    const int memlen = lengths[b] - 1;   // rows [0, memlen) are memory
    const int s0 = ch * ROWS_PER_CH;
    int s1 = s0 + ROWS_PER_CH;
    if (s1 > memlen) s1 = memlen;
    float acc = 0.0f;
    const float* p = sent + (size_t)b * SS * DD + (size_t)s0 * DD + d;
    for (int s = s0; s < s1; ++s) {      // coalesced: wave reads 128B/row
        acc += *p;
        p += DD;
    }
    partials[((size_t)b * CH + ch) * DD + d] = acc;
}

// ---------------------------------------------------------------------------
// Kernel 3: reduce partials -> x[b][d]
// ---------------------------------------------------------------------------
__global__ void k_reduce(const float* __restrict__ partials,
                         float* __restrict__ x) {
    const int b = blockIdx.x;
    const int d = threadIdx.x;
    float acc = 0.0f;
    #pragma unroll
    for (int c = 0; c < CH; ++c)
        acc += partials[((size_t)b * CH + c) * DD + d];
    x[(size_t)b * DD + d] = acc;
}

// ---------------------------------------------------------------------------
// Kernel 4: Y = x @ W via V_WMMA_F32_16X16X4_F32, then
//           out[b,d] = q0[b,d] + hops * Y[b,d]  (repeated add = exact match)
// One wave (32 lanes) per 16x16 output tile; grid = (DD/16, BB/16).
// ---------------------------------------------------------------------------
__global__ void k_gemm_wmma(const float* __restrict__ x,
                            const float* __restrict__ W,
                            const float* __restrict__ sent,
                            const int* __restrict__ lengths,
                            const int* __restrict__ hops_p,
                            float* __restrict__ out) {
    const int tn   = blockIdx.x;   // N tile (0..15)
    const int tm   = blockIdx.y;   // M tile (0..3)
    const int lane = threadIdx.x;  // 0..31 (wave32)
    const int half = lane >> 4;    // 0: K pair {0,1}; 1: K pair {2,3}
    const int lm   = lane & 15;

    v8f acc = {};
    const float* arow = x + (size_t)(tm * 16 + lm) * DD;

    for (int k = 0; k < DD; k += 4) {
        // A 16x4 f32 layout: lanes 0-15 -> K={k,k+1}, lanes 16-31 -> K={k+2,k+3}
        v2f a;
        a.x = arow[k + 2 * half];
        a.y = arow[k + 2 * half + 1];
        // B 4x16 f32 layout (rows striped across lanes, K split by lane half)
        v2f bm;
        bm.x = W[(size_t)(k + 2 * half)     * DD + tn * 16 + lm];
        bm.y = W[(size_t)(k + 2 * half + 1) * DD + tn * 16 + lm];
        acc = __builtin_amdgcn_wmma_f32_16x16x4_f32(
            /*neg_a=*/false, a, /*neg_b=*/false, bm,
            /*c_mod=*/(short)0, acc, /*reuse_a=*/false, /*reuse_b=*/false);
    }

    const int hops = *hops_p;
    // C/D 16x16 f32 layout: VGPR r -> row r (lanes 0-15) / row r+8 (lanes 16-31)
    #pragma unroll
    for (int r = 0; r < 8; ++r) {
        const int m = tm * 16 + r + 8 * half;   // global row == batch index b
        const int n = tn * 16 + lm;             // global col == feature d
        const int qrow = lengths[m] - 1;        // last valid slot
        float q = sent[((size_t)m * SS + qrow) * DD + n];
        const float y = acc[r];
        for (int h = 0; h < hops; ++h) q += y;  // matches reference rounding
        out[(size_t)m * DD + n] = q;
    }
}

// ---------------------------------------------------------------------------
// Launcher
// ---------------------------------------------------------------------------
extern "C" void kernel_launch(void* const* d_in, const int* in_sizes, int n_in,
                              void* d_out, int out_size, void* d_ws, size_t ws_size,
                              hipStream_t stream) {
    const float*         sent = (const float*)d_in[0];          // [B,S,D] f32
    const unsigned char* mask = (const unsigned char*)d_in[1];  // [B,S] bool
    const float*         W    = (const float*)d_in[2];          // [D,D] f32
    const int*           hops = (const int*)d_in[3];            // scalar int
    float*               out  = (float*)d_out;                  // [B,1,D] f32

    char* ws = (char*)d_ws;
    int*   lengths  = (int*)ws;                                       // 256 B
    float* partials = (float*)(ws + 256);                             // B*CH*D f32 = 1 MB
    float* x        = (float*)(ws + 256 + (size_t)BB * CH * DD * 4);  // B*D f32 = 64 KB

    k_lengths<<<BB, 256, 0, stream>>>(mask, lengths);
    k_partial<<<dim3(CH, BB), 256, 0, stream>>>(sent, lengths, partials);
    k_reduce<<<BB, 256, 0, stream>>>(partials, x);
    k_gemm_wmma<<<dim3(DD / 16, BB / 16), 32, 0, stream>>>(x, W, sent, lengths,
                                                           hops, out);
}